// GraphAttentionLayer_36532991820102
// MI455X (gfx1250) — compile-verified
//
#include <hip/hip_runtime.h>

typedef float v2f __attribute__((ext_vector_type(2)));
typedef float v4f __attribute__((ext_vector_type(4)));
typedef float v8f __attribute__((ext_vector_type(8)));

#define BB 16
#define NN 1024
#define FF 256
#define NEG_INF_V (-1000000000.0f)
#define SLOPE_V 0.2f

#define K3_WAVES 4            // waves per workgroup in kernel 3
#define BSTRIDE 20            // padded LDS row stride (floats): bank-conflict-free

// ---------------------------------------------------------------------------
// Kernel 1: WhT[b,o,n] = sum_k h[b,n,k] * W[k,o]   (Wh stored TRANSPOSED:
// o-major, n-innermost, so kernel 3's B-operand K-pairs are contiguous).
// One wave per 16x16 tile via v_wmma_f32_16x16x4_f32 (full fp32 precision).
// ---------------------------------------------------------------------------
__global__ __launch_bounds__(32)
void gat_gemm_hW(const float* __restrict__ h, const float* __restrict__ W,
                 float* __restrict__ WhT) {
  int tile = blockIdx.x;                  // B*(N/16)*(F/16) tiles
  int ot = tile % (FF / 16);
  int nt = (tile / (FF / 16)) % (NN / 16);
  int b  = tile / ((FF / 16) * (NN / 16));
  int lane = threadIdx.x;
  int n  = lane & 15;
  int hh = lane >> 4;

  const float* Arow = h + ((size_t)(b * NN + nt * 16 + n)) * FF;
  const float* Bcol = W + ot * 16 + n;

  v8f acc = {};
  #pragma unroll 4
  for (int k = 0; k < FF; k += 4) {
    v2f a = *(const v2f*)(Arow + k + 2 * hh);
    v2f bm;
    bm.x = Bcol[(size_t)(k + 2 * hh) * FF];
    bm.y = Bcol[(size_t)(k + 1 + 2 * hh) * FF];
    acc = __builtin_amdgcn_wmma_f32_16x16x4_f32(false, a, false, bm,
                                                (short)0, acc, false, false);
  }
  // lane holds rows (n-dim) r+8*hh at o-column ot*16+n: contiguous in WhT.
  float* outp = WhT + ((size_t)(b * FF + ot * 16 + n)) * NN + nt * 16 + 8 * hh;
  v4f lo = { acc[0], acc[1], acc[2], acc[3] };
  v4f hi = { acc[4], acc[5], acc[6], acc[7] };
  *(v4f*)(outp)     = lo;
  *(v4f*)(outp + 4) = hi;
}

// ---------------------------------------------------------------------------
// Kernel 2: f1[b,n] = sum_o WhT[b,o,n]*a[o],  f2 likewise with a[256:512].
// ---------------------------------------------------------------------------
__global__ __launch_bounds__(256)
void gat_f12(const float* __restrict__ WhT, const float* __restrict__ a,
             float* __restrict__ f1, float* __restrict__ f2) {
  int idx = blockIdx.x * blockDim.x + threadIdx.x;   // 0..B*N-1
  int b = idx / NN;
  int n = idx % NN;
  const float* base = WhT + (size_t)b * FF * NN + n;
  float s1 = 0.f, s2 = 0.f;
  #pragma unroll 8
  for (int o = 0; o < FF; ++o) {
    float w = base[(size_t)o * NN];
    s1 += w * a[o];
    s2 += w * a[FF + o];
  }
  f1[idx] = s1;
  f2[idx] = s2;
}

// ---------------------------------------------------------------------------
// Kernel 3: fused mask + online-softmax + (att @ Wh), flash-attention style.
// 4 waves per block, all sharing batch b and the jt stream: the 16KB
// WhT[:, j0:j0+16] B-panel is staged into LDS ONCE per block per j-step and
// consumed by all 4 waves (4x less L2 traffic than private streams).
// Each wave owns a 16-row i-tile with the full 16x256 fp32 accumulator in
// 16 x v8f. LDS rows padded to 20 floats -> conflict-free ds_load_b64.
// ---------------------------------------------------------------------------
__global__ __launch_bounds__(128)
void gat_attn(const float* __restrict__ WhT, const int* __restrict__ adj,
              const float* __restrict__ f1, const float* __restrict__ f2,
              float* __restrict__ out) {
  __shared__ float Btile[FF * BSTRIDE];                 // 20 KB, o-major panel
  __shared__ float ptile[K3_WAVES][16 * BSTRIDE];       // per-wave P tiles
  int tid  = threadIdx.x;
  int w    = tid >> 5;
  int lane = tid & 31;
  int n  = lane & 15;
  int hh = lane >> 4;
  int b     = blockIdx.x >> 4;                          // 16 supertiles per b
  int super = blockIdx.x & 15;
  int i0 = (super * K3_WAVES + w) * 16;

  // f1 for the 8 rows (r + 8*hh) this lane covers
  float f1v[8];
  {
    const float* p = f1 + b * NN + i0 + 8 * hh;
    v4f t0 = *(const v4f*)(p);
    v4f t1 = *(const v4f*)(p + 4);
    f1v[0] = t0.x; f1v[1] = t0.y; f1v[2] = t0.z; f1v[3] = t0.w;
    f1v[4] = t1.x; f1v[5] = t1.y; f1v[6] = t1.z; f1v[7] = t1.w;
  }

  v8f acc[16];
  v8f zero = {};
  #pragma unroll
  for (int t = 0; t < 16; ++t) acc[t] = zero;
  float mrun[8], lrun[8];
  #pragma unroll
  for (int r = 0; r < 8; ++r) { mrun[r] = -3.0e38f; lrun[r] = 0.f; }

  const int*   adjb = adj + ((size_t)(b * NN + i0 + 8 * hh)) * NN;
  const float* WhTb = WhT + (size_t)b * FF * NN;
  const float* f2b  = f2 + b * NN;

  for (int jt = 0; jt < NN / 16; ++jt) {
    int j0 = jt * 16;

    // ---- cooperative staging: WhT[b, o, j0:j0+16] for o = tid, tid+128 ----
    {
      const float* g0 = WhTb + (size_t)tid * NN + j0;
      const float* g1 = g0 + (size_t)128 * NN;
      v4f x0 = *(const v4f*)(g0);
      v4f x1 = *(const v4f*)(g0 + 4);
      v4f x2 = *(const v4f*)(g0 + 8);
      v4f x3 = *(const v4f*)(g0 + 12);
      v4f y0 = *(const v4f*)(g1);
      v4f y1 = *(const v4f*)(g1 + 4);
      v4f y2 = *(const v4f*)(g1 + 8);
      v4f y3 = *(const v4f*)(g1 + 12);
      float* l0 = Btile + tid * BSTRIDE;
      float* l1 = Btile + (tid + 128) * BSTRIDE;
      *(v4f*)(l0)      = x0;  *(v4f*)(l0 + 4)  = x1;
      *(v4f*)(l0 + 8)  = x2;  *(v4f*)(l0 + 12) = x3;
      *(v4f*)(l1)      = y0;  *(v4f*)(l1 + 4)  = y1;
      *(v4f*)(l1 + 8)  = y2;  *(v4f*)(l1 + 12) = y3;
    }

    // ---- scores + online softmax (per wave) ----
    float f2v = f2b[j0 + n];
    float p[8];
    #pragma unroll
    for (int r = 0; r < 8; ++r) {
      float e = f1v[r] + f2v;
      e = e > 0.f ? e : SLOPE_V * e;
      int av = adjb[(size_t)r * NN + j0 + n];
      e = (av == 0) ? NEG_INF_V : e;
      float rm = e;
      rm = fmaxf(rm, __shfl_xor(rm, 1, 32));
      rm = fmaxf(rm, __shfl_xor(rm, 2, 32));
      rm = fmaxf(rm, __shfl_xor(rm, 4, 32));
      rm = fmaxf(rm, __shfl_xor(rm, 8, 32));
      float nm   = fmaxf(mrun[r], rm);
      float pe   = __expf(e - nm);
      float corr = __expf(mrun[r] - nm);
      mrun[r] = nm;
      float rs = pe;
      rs += __shfl_xor(rs, 1, 32);
      rs += __shfl_xor(rs, 2, 32);
      rs += __shfl_xor(rs, 4, 32);
      rs += __shfl_xor(rs, 8, 32);
      lrun[r] = lrun[r] * corr + rs;
      p[r] = pe;
      #pragma unroll
      for (int t = 0; t < 16; ++t) acc[t][r] *= corr;   // online-softmax rescale
    }
    #pragma unroll
    for (int r = 0; r < 8; ++r)
      ptile[w][(r + 8 * hh) * BSTRIDE + n] = p[r];

    __syncthreads();   // Btile + ptile visible to consumers

    // A slices from own P tile: VGPR0 = P[M=n][K=4kk+2h], VGPR1 = K+1
    v2f A[4];
    #pragma unroll
    for (int kk = 0; kk < 4; ++kk)
      A[kk] = *(const v2f*)(&ptile[w][n * BSTRIDE + 4 * kk + 2 * hh]);

    // acc[t] += P(16x16) @ Wh[j0:j0+16, 16t:16t+16] ; B pairs from LDS panel
    #pragma unroll
    for (int t = 0; t < 16; ++t) {
      const float* Bp = Btile + (t * 16 + n) * BSTRIDE + 2 * hh;
      #pragma unroll
      for (int kk = 0; kk < 4; ++kk) {
        v2f bm = *(const v2f*)(Bp + 4 * kk);
        acc[t] = __builtin_amdgcn_wmma_f32_16x16x4_f32(false, A[kk], false, bm,
                                                       (short)0, acc[t],
                                                       false, false);
      }
    }
    __syncthreads();   // protect LDS tiles before next iteration's stores
  }

  float inv[8];
  #pragma unroll
  for (int r = 0; r < 8; ++r) inv[r] = 1.0f / lrun[r];
  float* ob = out + ((size_t)(b * NN + i0 + 8 * hh)) * FF + n;
  #pragma unroll
  for (int t = 0; t < 16; ++t) {
    #pragma unroll
    for (int r = 0; r < 8; ++r)
      ob[(size_t)r * FF + t * 16] = acc[t][r] * inv[r];
  }
}

extern "C" void kernel_launch(void* const* d_in, const int* in_sizes, int n_in,
                              void* d_out, int out_size, void* d_ws, size_t ws_size,
                              hipStream_t stream) {
  const float* h   = (const float*)d_in[0];
  const int*   adj = (const int*)d_in[1];
  const float* W   = (const float*)d_in[2];
  const float* a   = (const float*)d_in[3];
  float* out = (float*)d_out;

  float* WhT = (float*)d_ws;                      // 16 MB (transposed Wh)
  float* f1  = WhT + (size_t)BB * NN * FF;        // 64 KB
  float* f2  = f1 + (size_t)BB * NN;              // 64 KB

  gat_gemm_hW<<<BB * (NN / 16) * (FF / 16), 32, 0, stream>>>(h, W, WhT);
  gat_f12<<<(BB * NN) / 256, 256, 0, stream>>>(WhT, a, f1, f2);
  gat_attn<<<BB * (NN / 16 / K3_WAVES), 128, 0, stream>>>(WhT, adj, f1, f2, out);
}